// SpectralDropout_37254546325874
// MI455X (gfx1250) — compile-verified
//
#include <hip/hip_runtime.h>

// SpectralDropout: out[b,s,c] = x[b,s,c] * (feature_rand[s] >= 0.2f)
// Shapes: x (128, 128, 2048) fp32, feature_rand (128,) fp32.
// Pure streaming op: 268 MB traffic, ~33.5 MFLOP -> HBM-bound (~11.5 us floor
// at 23.3 TB/s). Optimal path on MI455X: wave32, b128 global ops, NT hints.

#define B_DIM 128
#define S_DIM 128
#define C_DIM 2048
#define DROP_P 0.2f

typedef __attribute__((ext_vector_type(4))) float floatx4;

__global__ __launch_bounds__(256) void spectral_dropout_b128_nt(
    const float* __restrict__ x,
    const float* __restrict__ feature_rand,
    float* __restrict__ out)
{
    // blockIdx.x = spectral feature s (uniform per block -> scalar mask load)
    // blockIdx.y = batch b
    const int s = blockIdx.x;
    const int b = blockIdx.y;

    // Block-uniform: compiler lowers this to s_load_b32 + scalar compare.
    const float scale = (feature_rand[s] >= DROP_P) ? 1.0f : 0.0f;

    const size_t row = ((size_t)b * S_DIM + s) * (size_t)C_DIM;
    const floatx4* __restrict__ xi = reinterpret_cast<const floatx4*>(x + row);
    floatx4* __restrict__ oi       = reinterpret_cast<floatx4*>(out + row);

    // 2048 floats / 4 = 512 float4 per row; 256 threads * 2 float4 each.
    const int t = threadIdx.x;

    // Streamed once, never reused; footprint (256 MB) > L2 (192 MB):
    // non-temporal load/store keeps the stream from thrashing L2.
    floatx4 v0 = __builtin_nontemporal_load(&xi[t]);
    floatx4 v1 = __builtin_nontemporal_load(&xi[t + 256]);

    v0 *= scale;
    v1 *= scale;

    __builtin_nontemporal_store(v0, &oi[t]);
    __builtin_nontemporal_store(v1, &oi[t + 256]);
}

extern "C" void kernel_launch(void* const* d_in, const int* in_sizes, int n_in,
                              void* d_out, int out_size, void* d_ws, size_t ws_size,
                              hipStream_t stream) {
    (void)in_sizes; (void)n_in; (void)d_ws; (void)ws_size; (void)out_size;

    const float* x            = (const float*)d_in[0];  // (128,128,2048) fp32
    const float* feature_rand = (const float*)d_in[1];  // (128,) fp32
    float* out                = (float*)d_out;

    dim3 grid(S_DIM, B_DIM);   // one block per (s, b) row
    dim3 block(256);           // 8 wave32 waves per block
    spectral_dropout_b128_nt<<<grid, block, 0, stream>>>(x, feature_rand, out);
}